// ICP_15839839387875
// MI455X (gfx1250) — compile-verified
//
#include <hip/hip_runtime.h>
#include <math.h>

#define BB 4
#define NN 4096
#define NTILE 256   /* NN/16 */
#define MTILE 256   /* NN/16 */
#define ICP_STEPS 16
#define ICP_TOL 1e-6f

typedef float v2f __attribute__((ext_vector_type(2)));
typedef float v8f __attribute__((ext_vector_type(8)));

// ---------------------------------------------------------------------------
// prep: temporal = source; pack target into WMMA B-matrix tile layout:
//   B column m = [bx, by, bz, |b|^2]; lanes 0-15 hold (K=0,K=1)=(x,y) of N=lane,
//   lanes 16-31 hold (K=2,K=3)=(z,|b|^2) of N=lane-16. Also zero errlast/done.
// ---------------------------------------------------------------------------
__global__ void icp_prep(const float* __restrict__ src,
                         const float* __restrict__ tgt,
                         float* __restrict__ temporal,
                         float2* __restrict__ bpack,
                         float* __restrict__ errlast,
                         int* __restrict__ done)
{
    int t = blockIdx.x * blockDim.x + threadIdx.x;
    if (t < BB * NN) {
        temporal[t * 3 + 0] = src[t * 3 + 0];
        temporal[t * 3 + 1] = src[t * 3 + 1];
        temporal[t * 3 + 2] = src[t * 3 + 2];
    }
    if (t < BB * MTILE * 32) {
        int b    = t / (MTILE * 32);
        int r    = t % (MTILE * 32);
        int mt   = r >> 5;
        int lane = r & 31;
        int m    = mt * 16 + (lane & 15);
        const float* p = tgt + (b * NN + m) * 3;
        float x = p[0], y = p[1], z = p[2];
        float2 v;
        if (lane < 16) { v.x = x; v.y = y; }
        else           { v.x = z; v.y = x * x + y * y + z * z; }
        bpack[t] = v;
    }
    if (t < BB) errlast[t] = 0.0f;
    if (t == 0) *done = 0;
}

// ---------------------------------------------------------------------------
// NN search: one wave per (batch, 16-row source tile). Inner loop over 256
// target tiles: load pre-packed B tile, V_WMMA_F32_16X16X4_F32 produces the
// 16x16 tile of (|b|^2 - 2 a.b); running per-lane min/argmin, then a 16-lane
// shuffle reduction over the N dimension of the C layout.
// A row (source point) = [-2x, -2y, -2z, 1]:
//   lanes 0-15: VGPR0=K0=-2x, VGPR1=K1=-2y (M=lane)
//   lanes16-31: VGPR0=K2=-2z, VGPR1=K3= 1  (M=lane-16)
// ---------------------------------------------------------------------------
__global__ void icp_nn(const float* __restrict__ temporal,
                       const float2* __restrict__ bpack,
                       float* __restrict__ bestval,
                       int* __restrict__ bestidx)
{
    int wave  = (blockIdx.x * blockDim.x + threadIdx.x) >> 5;
    int lane  = threadIdx.x & 31;
    int b     = wave >> 8;     // / NTILE
    int nt    = wave & 255;    // % NTILE
    int laneM = lane & 15;

    const float* sp = temporal + (b * NN + nt * 16 + laneM) * 3;
    float x = sp[0], y = sp[1], z = sp[2];
    v2f A;
    if (lane < 16) { A.x = -2.0f * x; A.y = -2.0f * y; }
    else           { A.x = -2.0f * z; A.y = 1.0f;      }

    float rv[8];
    int   ri[8];
#pragma unroll
    for (int v = 0; v < 8; ++v) { rv[v] = 3.0e38f; ri[v] = 0; }

    const float2* bp = bpack + (size_t)(b * MTILE) * 32 + lane;
#pragma unroll 4
    for (int mt = 0; mt < MTILE; ++mt) {
        float2 bv = bp[mt * 32];
        if (mt + 1 < MTILE)
            __builtin_prefetch((const void*)(bp + (mt + 1) * 32), 0, 1);
        v2f Bv; Bv.x = bv.x; Bv.y = bv.y;
        v8f acc = {};
        acc = __builtin_amdgcn_wmma_f32_16x16x4_f32(false, A, false, Bv,
                                                    (short)0, acc, false, false);
        int mc = mt * 16 + laneM;   // this lane's N-column within the tile
#pragma unroll
        for (int v = 0; v < 8; ++v) {
            float d = acc[v];
            if (d < rv[v]) { rv[v] = d; ri[v] = mc; }
        }
    }

    // reduce over the 16 N-lanes; xor masks < 16 never cross half-waves
#pragma unroll
    for (int off = 8; off >= 1; off >>= 1) {
#pragma unroll
        for (int v = 0; v < 8; ++v) {
            float ov = __shfl_xor(rv[v], off, 32);
            int   oi = __shfl_xor(ri[v], off, 32);
            if (ov < rv[v] || (ov == rv[v] && oi < ri[v])) { rv[v] = ov; ri[v] = oi; }
        }
    }

    // C layout: VGPR v -> M=v (lanes 0-15), M=v+8 (lanes 16-31)
    if (laneM == 0) {
        int mrow0 = (lane >> 4) * 8;
#pragma unroll
        for (int v = 0; v < 8; ++v) {
            int n = nt * 16 + mrow0 + v;
            bestval[b * NN + n] = rv[v];
            bestidx[b * NN + n] = ri[v];
        }
    }
}

// ---------------------------------------------------------------------------
// per-batch fixed-order reduction of Kabsch sums:
//   sums[0..2]=sum t, [3..5]=sum k, [6..14]=sum t_i k_j, [15]=sum sqrt(d2)
// use_nn=1: k = target[bestidx[n]], d2 = |t|^2 + bestval (NN iteration)
// use_nn=0: k = Bpts[n] (final source-vs-temporal Kabsch), no error term
// ---------------------------------------------------------------------------
__global__ void icp_reduce(const float* __restrict__ Apts,
                           const float* __restrict__ Bpts,
                           const int* __restrict__ bestidx,
                           const float* __restrict__ bestval,
                           float* __restrict__ sums,
                           int use_nn)
{
    __shared__ float part[256][16];
    int b   = blockIdx.x;
    int tid = threadIdx.x;
    float acc[16];
#pragma unroll
    for (int q = 0; q < 16; ++q) acc[q] = 0.0f;

    for (int n = tid; n < NN; n += 256) {
        const float* ap = Apts + (b * NN + n) * 3;
        float ax = ap[0], ay = ap[1], az = ap[2];
        float kx, ky, kz;
        if (use_nn) {
            int idx = bestidx[b * NN + n];
            const float* kp = Bpts + (b * NN + idx) * 3;
            kx = kp[0]; ky = kp[1]; kz = kp[2];
            float a2 = ax * ax + ay * ay + az * az;
            float d2 = a2 + bestval[b * NN + n];
            acc[15] += sqrtf(fmaxf(d2, 0.0f));
        } else {
            const float* kp = Bpts + (b * NN + n) * 3;
            kx = kp[0]; ky = kp[1]; kz = kp[2];
        }
        acc[0] += ax; acc[1] += ay; acc[2] += az;
        acc[3] += kx; acc[4] += ky; acc[5] += kz;
        acc[6] += ax * kx; acc[7]  += ax * ky; acc[8]  += ax * kz;
        acc[9] += ay * kx; acc[10] += ay * ky; acc[11] += ay * kz;
        acc[12]+= az * kx; acc[13] += az * ky; acc[14] += az * kz;
    }
#pragma unroll
    for (int q = 0; q < 16; ++q) part[tid][q] = acc[q];
    __syncthreads();
    if (tid < 16) {
        float s = 0.0f;
        for (int i = 0; i < 256; ++i) s += part[i][tid];   // fixed order
        sums[b * 16 + tid] = s;
    }
}

// ---------------------------------------------------------------------------
// 3x3 Kabsch rotation from centered cross-covariance H = sum (t-cs)(k-ct)^T.
// Jacobi eigensolve of H^T H -> V; U cols = H v_i / s_i (u2 = u0 x u1);
// R = V diag(1,1,det(V U^T)) U^T.
// ---------------------------------------------------------------------------
__device__ void kabsch3x3(const float H[9], float R[9])
{
    float Kc[9];
    for (int i = 0; i < 3; ++i)
        for (int j = 0; j < 3; ++j) {
            float s = 0.0f;
            for (int l = 0; l < 3; ++l) s += H[l * 3 + i] * H[l * 3 + j];
            Kc[i * 3 + j] = s;
        }
    float V[9]  = {1,0,0, 0,1,0, 0,0,1};
    float Am[9];
    for (int i = 0; i < 9; ++i) Am[i] = Kc[i];
    for (int sweep = 0; sweep < 12; ++sweep) {
        for (int pair = 0; pair < 3; ++pair) {
            int p = (pair == 2) ? 1 : 0;
            int q = (pair == 0) ? 1 : 2;
            float apq = Am[p * 3 + q];
            if (fabsf(apq) < 1e-30f) continue;
            float app = Am[p * 3 + p], aqq = Am[q * 3 + q];
            float tau = (aqq - app) / (2.0f * apq);
            float tt  = (tau >= 0.0f ? 1.0f : -1.0f) /
                        (fabsf(tau) + sqrtf(1.0f + tau * tau));
            float c = 1.0f / sqrtf(1.0f + tt * tt);
            float s = tt * c;
            for (int k = 0; k < 3; ++k) {
                float akp = Am[k * 3 + p], akq = Am[k * 3 + q];
                Am[k * 3 + p] = c * akp - s * akq;
                Am[k * 3 + q] = s * akp + c * akq;
            }
            for (int k = 0; k < 3; ++k) {
                float apk = Am[p * 3 + k], aqk = Am[q * 3 + k];
                Am[p * 3 + k] = c * apk - s * aqk;
                Am[q * 3 + k] = s * apk + c * aqk;
            }
            for (int k = 0; k < 3; ++k) {
                float vkp = V[k * 3 + p], vkq = V[k * 3 + q];
                V[k * 3 + p] = c * vkp - s * vkq;
                V[k * 3 + q] = s * vkp + c * vkq;
            }
        }
    }
    float w[3] = {Am[0], Am[4], Am[8]};
    int ord[3] = {0, 1, 2};
    for (int i = 0; i < 2; ++i)
        for (int j = 0; j < 2 - i; ++j)
            if (w[ord[j]] < w[ord[j + 1]]) { int o = ord[j]; ord[j] = ord[j + 1]; ord[j + 1] = o; }
    float Vs[9];
    for (int k = 0; k < 3; ++k)
        for (int i = 0; i < 3; ++i)
            Vs[i * 3 + k] = V[i * 3 + ord[k]];

    float u[3][3];
    float hv[3];
    for (int i = 0; i < 3; ++i)
        hv[i] = H[i*3+0]*Vs[0*3+0] + H[i*3+1]*Vs[1*3+0] + H[i*3+2]*Vs[2*3+0];
    float n0 = sqrtf(hv[0]*hv[0] + hv[1]*hv[1] + hv[2]*hv[2]);
    if (n0 > 1e-12f) { u[0][0]=hv[0]/n0; u[0][1]=hv[1]/n0; u[0][2]=hv[2]/n0; }
    else             { u[0][0]=1.0f; u[0][1]=0.0f; u[0][2]=0.0f; }
    for (int i = 0; i < 3; ++i)
        hv[i] = H[i*3+0]*Vs[0*3+1] + H[i*3+1]*Vs[1*3+1] + H[i*3+2]*Vs[2*3+1];
    float dp = hv[0]*u[0][0] + hv[1]*u[0][1] + hv[2]*u[0][2];
    for (int i = 0; i < 3; ++i) hv[i] -= dp * u[0][i];
    float n1 = sqrtf(hv[0]*hv[0] + hv[1]*hv[1] + hv[2]*hv[2]);
    if (n1 > 1e-12f) { u[1][0]=hv[0]/n1; u[1][1]=hv[1]/n1; u[1][2]=hv[2]/n1; }
    else {
        float ax = fabsf(u[0][0]);
        float t0[3] = {(ax < 0.9f) ? 1.0f : 0.0f, (ax < 0.9f) ? 0.0f : 1.0f, 0.0f};
        float d2 = t0[0]*u[0][0] + t0[1]*u[0][1] + t0[2]*u[0][2];
        for (int i = 0; i < 3; ++i) hv[i] = t0[i] - d2 * u[0][i];
        float nn2 = sqrtf(hv[0]*hv[0] + hv[1]*hv[1] + hv[2]*hv[2]);
        u[1][0]=hv[0]/nn2; u[1][1]=hv[1]/nn2; u[1][2]=hv[2]/nn2;
    }
    u[2][0] = u[0][1]*u[1][2] - u[0][2]*u[1][1];
    u[2][1] = u[0][2]*u[1][0] - u[0][0]*u[1][2];
    u[2][2] = u[0][0]*u[1][1] - u[0][1]*u[1][0];

    float Mt[9];
    for (int i = 0; i < 3; ++i)
        for (int j = 0; j < 3; ++j)
            Mt[i*3+j] = Vs[i*3+0]*u[0][j] + Vs[i*3+1]*u[1][j] + Vs[i*3+2]*u[2][j];
    float det = Mt[0]*(Mt[4]*Mt[8]-Mt[5]*Mt[7])
              - Mt[1]*(Mt[3]*Mt[8]-Mt[5]*Mt[6])
              + Mt[2]*(Mt[3]*Mt[7]-Mt[4]*Mt[6]);
    float dfix = (det < 0.0f) ? -1.0f : 1.0f;
    for (int i = 0; i < 3; ++i)
        for (int j = 0; j < 3; ++j)
            R[i*3+j] = Mt[i*3+j] + (dfix - 1.0f) * Vs[i*3+2] * u[2][j];
}

// ---------------------------------------------------------------------------
// per-iteration solve: errnew, scalar all-batch done flag, R|t per batch
// ---------------------------------------------------------------------------
__global__ void icp_solve(const float* __restrict__ sums,
                          float* __restrict__ Rt,
                          float* __restrict__ errlast,
                          int* __restrict__ done)
{
    __shared__ int conv[BB];
    __shared__ int dnew;
    int b = threadIdx.x;
    float st[3], sk[3], Hs[9], errnew = 0.0f;
    if (b < BB) {
        const float* s = sums + b * 16;
        st[0]=s[0]; st[1]=s[1]; st[2]=s[2];
        sk[0]=s[3]; sk[1]=s[4]; sk[2]=s[5];
        for (int q = 0; q < 9; ++q) Hs[q] = s[6 + q];
        errnew = s[15] / (float)NN;
        conv[b] = (fabsf(errnew - errlast[b]) < ICP_TOL) ? 1 : 0;
    }
    __syncthreads();
    if (threadIdx.x == 0) {
        int all = 1;
        for (int i = 0; i < BB; ++i) all &= conv[i];
        dnew = (*done) | all;
        *done = dnew;
    }
    __syncthreads();
    if (b < BB && !dnew) {
        float cs[3], ct[3];
        for (int i = 0; i < 3; ++i) { cs[i] = st[i] / (float)NN; ct[i] = sk[i] / (float)NN; }
        float Hc[9];
        for (int i = 0; i < 3; ++i)
            for (int j = 0; j < 3; ++j)
                Hc[i*3+j] = Hs[i*3+j] - (float)NN * cs[i] * ct[j];
        float R[9];
        kabsch3x3(Hc, R);
        float* rt = Rt + b * 12;
        for (int q = 0; q < 9; ++q) rt[q] = R[q];
        for (int i = 0; i < 3; ++i)
            rt[9 + i] = ct[i] - (R[i*3+0]*cs[0] + R[i*3+1]*cs[1] + R[i*3+2]*cs[2]);
        errlast[b] = errnew;
    }
}

__global__ void icp_apply(float* __restrict__ temporal,
                          const float* __restrict__ Rt,
                          const int* __restrict__ done)
{
    if (*done) return;
    int t = blockIdx.x * blockDim.x + threadIdx.x;
    if (t >= BB * NN) return;
    int b = t / NN;
    const float* rt = Rt + b * 12;
    float* p = temporal + t * 3;
    float x = p[0], y = p[1], z = p[2];
    float nx = rt[0]*x + rt[1]*y + rt[2]*z + rt[9];
    float ny = rt[3]*x + rt[4]*y + rt[5]*z + rt[10];
    float nz = rt[6]*x + rt[7]*y + rt[8]*z + rt[11];
    p[0] = nx; p[1] = ny; p[2] = nz;
}

__global__ void icp_final(const float* __restrict__ sums, float* __restrict__ out)
{
    int b = threadIdx.x;
    if (b >= BB) return;
    const float* s = sums + b * 16;
    float cs[3] = {s[0]/(float)NN, s[1]/(float)NN, s[2]/(float)NN};
    float ct[3] = {s[3]/(float)NN, s[4]/(float)NN, s[5]/(float)NN};
    float Hc[9];
    for (int i = 0; i < 3; ++i)
        for (int j = 0; j < 3; ++j)
            Hc[i*3+j] = s[6 + i*3 + j] - (float)NN * cs[i] * ct[j];
    float R[9];
    kabsch3x3(Hc, R);
    for (int i = 0; i < 3; ++i) {
        out[b*12 + i*4 + 0] = R[i*3+0];
        out[b*12 + i*4 + 1] = R[i*3+1];
        out[b*12 + i*4 + 2] = R[i*3+2];
        out[b*12 + i*4 + 3] = ct[i] - (R[i*3+0]*cs[0] + R[i*3+1]*cs[1] + R[i*3+2]*cs[2]);
    }
}

// ---------------------------------------------------------------------------
extern "C" void kernel_launch(void* const* d_in, const int* in_sizes, int n_in,
                              void* d_out, int out_size, void* d_ws, size_t ws_size,
                              hipStream_t stream) {
    (void)in_sizes; (void)n_in; (void)out_size; (void)ws_size;
    const float* src = (const float*)d_in[0];
    const float* tgt = (const float*)d_in[1];
    float* out = (float*)d_out;

    char* ws = (char*)d_ws;
    const size_t OFF_TEMP  = 0;                                  // B*N*3 f
    const size_t OFF_BPACK = OFF_TEMP  + (size_t)BB*NN*3*4;      // B*MTILE*32 float2
    const size_t OFF_BVAL  = OFF_BPACK + (size_t)BB*MTILE*32*8;  // B*N f
    const size_t OFF_BIDX  = OFF_BVAL  + (size_t)BB*NN*4;        // B*N i
    const size_t OFF_SUMS  = OFF_BIDX  + (size_t)BB*NN*4;        // B*16 f
    const size_t OFF_RT    = OFF_SUMS  + (size_t)BB*16*4;        // B*12 f
    const size_t OFF_ERR   = OFF_RT    + (size_t)BB*12*4;        // B f
    const size_t OFF_DONE  = OFF_ERR   + (size_t)BB*4;           // 1 i

    float*  temporal = (float*) (ws + OFF_TEMP);
    float2* bpack    = (float2*)(ws + OFF_BPACK);
    float*  bestval  = (float*) (ws + OFF_BVAL);
    int*    bestidx  = (int*)   (ws + OFF_BIDX);
    float*  sums     = (float*) (ws + OFF_SUMS);
    float*  Rt       = (float*) (ws + OFF_RT);
    float*  errlast  = (float*) (ws + OFF_ERR);
    int*    done     = (int*)   (ws + OFF_DONE);

    icp_prep<<<128, 256, 0, stream>>>(src, tgt, temporal, bpack, errlast, done);

    for (int it = 0; it < ICP_STEPS; ++it) {
        icp_nn    <<<128, 256, 0, stream>>>(temporal, bpack, bestval, bestidx);
        icp_reduce<<<BB,  256, 0, stream>>>(temporal, tgt, bestidx, bestval, sums, 1);
        icp_solve <<<1,    32, 0, stream>>>(sums, Rt, errlast, done);
        icp_apply <<<64,  256, 0, stream>>>(temporal, Rt, done);
    }

    icp_reduce<<<BB, 256, 0, stream>>>(src, temporal, bestidx, bestval, sums, 0);
    icp_final <<<1,   32, 0, stream>>>(sums, out);
}